// GATLayer_32126355374743
// MI455X (gfx1250) — compile-verified
//
#include <hip/hip_runtime.h>
#include <hip/hip_bf16.h>
#include <math.h>

typedef __attribute__((ext_vector_type(2))) float v2f;
typedef __attribute__((ext_vector_type(8))) float v8f;

#define BATCH   2048
#define NNODE   128
#define FIN     256
#define FOUT    256
#define NEGVAL  (-9.0e15f)
#define BETA    0.2f

// ---------------------------------------------------------------------------
// Kernel A: wa1 = W @ a[:256], wa2 = W @ a[256:];  p1/p2 = 1000 * pos_table @ wa
// ws layout (floats): [0,256)=wa1  [256,512)=wa2  [512,640)=p1  [640,768)=p2
//                     [1024, 1024+262144)=s1  [+262144)=s2
// ---------------------------------------------------------------------------
__global__ __launch_bounds__(256) void gat_setup(const float* __restrict__ W,
                                                 const float* __restrict__ a,
                                                 float* __restrict__ ws) {
    __shared__ float lw1[FIN], lw2[FIN];
    int t = threadIdx.x;                       // 256 threads, one f each
    float acc1 = 0.f, acc2 = 0.f;
    for (int o = 0; o < FOUT; ++o) {
        float w = W[t * FOUT + o];
        acc1 += w * a[o];
        acc2 += w * a[FOUT + o];
    }
    ws[t]       = acc1;
    ws[256 + t] = acc2;
    lw1[t] = acc1;
    lw2[t] = acc2;
    __syncthreads();
    if (t < NNODE) {                           // one node position each
        float pa = 0.f, pb = 0.f;
        for (int f = 0; f < FIN; ++f) {
            float expo = (float)(f & ~1) * (1.0f / (float)FIN);  // 2*(f//2)/d
            float freq = __powf(10000.0f, -expo);
            float ang  = (float)t * freq;
            float pt   = (f & 1) ? __cosf(ang) : __sinf(ang);
            pa += pt * lw1[f];
            pb += pt * lw2[f];
        }
        ws[512 + t] = 1000.0f * pa;
        ws[640 + t] = 1000.0f * pb;
    }
}

// ---------------------------------------------------------------------------
// Kernel B: s1[b,n] = src[b,n,:]·wa1 + p1[n];  s2 likewise — via f32 WMMA.
// One wave per 16-row tile. B operand is branch-free: each lane reads its
// column of [wa1 | wa2 | zeros] from a per-lane LDS base, so the inner loop
// is exactly {global_load_b64, ds_load_b64, v_wmma} with no EXEC churn.
// ---------------------------------------------------------------------------
__global__ __launch_bounds__(256) void gat_scores(const float* __restrict__ src,
                                                  const float* __restrict__ ws,
                                                  float* __restrict__ s1,
                                                  float* __restrict__ s2) {
    // [0,256)=wa1  [256,512)=wa2  [512,768)=zeros  (+ p1/p2)
    __shared__ __align__(16) float lb[3 * FIN];
    __shared__ float lp1[NNODE], lp2[NNODE];
    int t = threadIdx.x;
    lb[t]           = ws[t];         // wa1
    lb[FIN + t]     = ws[FIN + t];   // wa2
    lb[2 * FIN + t] = 0.f;           // zero segment for lanes with n>=2
    if (t < NNODE) { lp1[t] = ws[512 + t]; lp2[t] = ws[640 + t]; }
    __syncthreads();

    int lane = t & 31;
    int wave = t >> 5;
    int tile = blockIdx.x * 8 + wave;          // 16 rows of (B*N, 256)
    int m    = lane & 15;                      // A row / D column index
    int hi   = lane >> 4;                      // half-wave
    int koff = hi << 1;                        // this half's K sub-offset

    const float* row  = src + (size_t)(tile * 16 + m) * FIN + koff;
    // per-lane B base: col0 -> wa1, col1 -> wa2, cols 2..15 -> zeros
    int bsel = (m == 0) ? 0 : ((m == 1) ? FIN : 2 * FIN);
    const float* bptr = lb + bsel + koff;

    v8f c = {0.f, 0.f, 0.f, 0.f, 0.f, 0.f, 0.f, 0.f};
    #pragma unroll 8
    for (int k = 0; k < FIN; k += 4) {
        if ((k & 31) == 0)                      // once per unrolled chunk:
            __builtin_prefetch(row + k + 64, 0, 3);  // global_prefetch_b8
        float2 al = *(const float2*)(row + k);  // A: K = k+koff, k+koff+1
        float2 bl = *(const float2*)(bptr + k); // B: same K, this lane's col
        v2f av;  av.x = al.x;  av.y = al.y;
        v2f bv;  bv.x = bl.x;  bv.y = bl.y;
        // D = A(16x4,f32) * B(4x16,f32) + C(16x16,f32)
        c = __builtin_amdgcn_wmma_f32_16x16x4_f32(false, av, false, bv,
                                                  (short)0, c, false, false);
    }

    if (m < 2) {                               // col0 -> s1, col1 -> s2
        float* so        = (m == 0) ? s1  : s2;
        const float* pp  = (m == 0) ? lp1 : lp2;
        int base = tile * 16 + hi * 8;         // D: lane<16 rows 0-7, else 8-15
        #pragma unroll
        for (int r = 0; r < 8; ++r) {
            int grow = base + r;
            so[grow] = c[r] + pp[grow & (NNODE - 1)];
        }
    }
}

// ---------------------------------------------------------------------------
// Kernel C: e[i,j] = leaky(s1[idxA] + s2[idxB]); mask by adj; softmax axis=1
// (over i, per column j). One block per batch; 128x128 e-tile in LDS (64KB).
// ---------------------------------------------------------------------------
__global__ __launch_bounds__(128) void gat_softmax(const float* __restrict__ adj,
                                                   const float* __restrict__ s1,
                                                   const float* __restrict__ s2,
                                                   float* __restrict__ out) {
    __shared__ float e_s[NNODE * NNODE];
    __shared__ float s1l[NNODE], s2l[NNODE];
    int t = threadIdx.x;                       // column j = t
    int b = blockIdx.x;
    const size_t bo = (size_t)b * (NNODE * NNODE);

    s1l[t] = s1[b * NNODE + t];
    s2l[t] = s2[b * NNODE + t];
    __syncthreads();

    for (int i = 0; i < NNODE; ++i) {
        int idx = i * NNODE + t;
        int r0 = idx << 1, r1 = r0 + 1;
        int iA = (r0 < NNODE * NNODE) ? (r0 >> 7) : (r0 & (NNODE - 1));
        int iB = (r1 < NNODE * NNODE) ? (r1 >> 7) : (r1 & (NNODE - 1));
        float v = s1l[iA] + s2l[iB];
        v = (v > 0.f) ? v : BETA * v;          // leaky relu
        float am = adj[bo + idx];              // coalesced
        e_s[idx] = (am > 0.f) ? v : NEGVAL;
    }
    __syncthreads();

    float mx = -INFINITY;
    for (int i = 0; i < NNODE; ++i)
        mx = fmaxf(mx, e_s[i * NNODE + t]);    // bank j%64: conflict-free
    float sum = 0.f;
    for (int i = 0; i < NNODE; ++i) {
        float v = __expf(e_s[i * NNODE + t] - mx);
        e_s[i * NNODE + t] = v;
        sum += v;
    }
    float inv = 1.0f / sum;
    for (int i = 0; i < NNODE; ++i)
        out[bo + i * NNODE + t] = e_s[i * NNODE + t] * inv;   // coalesced
}

// ---------------------------------------------------------------------------
extern "C" void kernel_launch(void* const* d_in, const int* in_sizes, int n_in,
                              void* d_out, int out_size, void* d_ws, size_t ws_size,
                              hipStream_t stream) {
    const float* src = (const float*)d_in[0];
    const float* W   = (const float*)d_in[1];
    const float* a   = (const float*)d_in[2];
    const float* adj = (const float*)d_in[3];
    float* out = (float*)d_out;
    float* ws  = (float*)d_ws;

    float* s1 = ws + 1024;
    float* s2 = ws + 1024 + (BATCH * NNODE);

    gat_setup  <<<1,    256, 0, stream>>>(W, a, ws);
    gat_scores <<<(BATCH * NNODE) / (16 * 8), 256, 0, stream>>>(src, ws, s1, s2);
    gat_softmax<<<BATCH, 128, 0, stream>>>(adj, s1, s2, out);
}